// CorrTorch_57715770523753
// MI455X (gfx1250) — compile-verified
//
#include <hip/hip_runtime.h>

// CDNA5 / gfx1250: correlation volume via V_WMMA_F32_16X16X4_F32 (fp32-exact path).
//
// One wave32 handles a 16-pixel W-tile for one (b, h) and all 81 displacements.
// A = fmap0 tile [16 pixels x 128 channels] (cached in 64 VGPRs).
// For each oy: B = fmap1 row y=h+oy-4, cols [w0-4 .. w0+19] as two 16-col N tiles.
// D = A x B; the needed outputs are the diagonal band D[m][m+ox], ox in [0,8].
//
// Interior W-tiles (14 of 16) take a fully unmasked load path; only the first
// and last tile use clamped-address + v_cndmask zero-select (no EXEC juggling).

typedef float v2f __attribute__((ext_vector_type(2)));
typedef float v8f __attribute__((ext_vector_type(8)));

#define BQ   4
#define CC   128
#define HH   128
#define WW   256
#define PS   9      // patch size (9x9 -> 81 displacement channels)
#define ND   (PS * PS)
#define PAD  4

__global__ __launch_bounds__(256) void corr_wmma_f32_kernel(
    const float* __restrict__ f0,
    const float* __restrict__ f1,
    float* __restrict__ out)
{
    const int lane      = threadIdx.x & 31;
    const int waveInBlk = threadIdx.x >> 5;
    const int tile      = blockIdx.x * 8 + waveInBlk;   // 8192 tiles total

    const int wt = tile & 15;            // W/16 = 16 tiles along W
    const int h  = (tile >> 4) & 127;    // H = 128
    const int b  = tile >> 11;           // B = 4
    const int w0 = wt * 16;

    const int hi = lane >> 4;            // half-wave: selects K half / M half
    const int ln = lane & 15;            // m for A rows, n for B/D columns

    // Per-wave LDS patch for output reshuffle: 9 rows x 17 floats (pad -> no bank conflicts)
    __shared__ float ldsBuf[8][PS * 17];
    float* myLds = ldsBuf[waveInBlk];

    const size_t planeHW = (size_t)HH * WW;                       // 32768
    const float* f0base = f0 + ((size_t)b * CC) * planeHW + (size_t)h * WW;
    const float* f1base = f1 + ((size_t)b * CC) * planeHW;

    // ---- Load A once: fp32 16x4 WMMA A-layout per K-step kk (K = 4*kk..4*kk+3):
    //   lane (hi, m): a[kk].x = f0[c=4kk+2hi][w0+m]   (VGPR0: K=0 | K=2)
    //                 a[kk].y = f0[c=4kk+2hi+1][w0+m] (VGPR1: K=1 | K=3)
    v2f a[32];
#pragma unroll
    for (int kk = 0; kk < 32; ++kk) {
        const float* p = f0base + (size_t)(4 * kk + 2 * hi) * planeHW + (w0 + ln);
        a[kk].x = p[0];
        a[kk].y = p[planeHW];
    }

    // Per-lane B columns for the two N tiles (global x in fmap1 coordinates).
    const int xA = w0 - PAD + ln;        // N-tile 0 column
    const int xB = xA + 16;              // N-tile 1 column
    const bool okA = (xA >= 0);          // xA < WW always (max 251)
    const bool okB = (xB < WW);          // xB >= 0 always (min 12)
    const int xAc = okA ? xA : 0;        // clamped in-bounds addresses
    const int xBc = okB ? xB : (WW - 1);
    const bool edgeTile = (wt == 0) | (wt == 15);

    for (int oy = 0; oy < PS; ++oy) {
        const int y = h + oy - PAD;
        float* outOy = out + (((size_t)b * ND + (size_t)oy * PS) * HH + (size_t)h) * WW + w0;

        // Prefetch next oy's fmap1 row (speculative; global_prefetch_b8)
        {
            int yn = y + 1; if (yn < 0) yn = 0; if (yn >= HH) yn = HH - 1;
            __builtin_prefetch(f1base + (size_t)yn * WW + (w0 + ln), 0, 1);
        }

        if (y < 0 || y >= HH) {
            // Zero-padded rows: output is exactly zero for all 9 ox at this oy.
            for (int e = lane; e < PS * 16; e += 32) {
                const int ox = e >> 4, m = e & 15;
                outOy[(size_t)ox * planeHW + m] = 0.0f;
            }
            continue;
        }

        const float* f1row = f1base + (size_t)y * WW;
        const float* pA = f1row + xAc;   // per-lane column pointers (clamped)
        const float* pB = f1row + xBc;

        v8f acc0 = {};
        v8f acc1 = {};

        if (!edgeTile) {
            // Interior tiles: all columns in-bounds, no masking at all.
#pragma unroll
            for (int kk = 0; kk < 32; ++kk) {
                const size_t co = (size_t)(4 * kk + 2 * hi) * planeHW;
                v2f b0, b1;
                b0.x = pA[co];           b0.y = pA[co + planeHW];
                b1.x = pB[co];           b1.y = pB[co + planeHW];
                acc0 = __builtin_amdgcn_wmma_f32_16x16x4_f32(
                    false, a[kk], false, b0, (short)0, acc0, false, false);
                acc1 = __builtin_amdgcn_wmma_f32_16x16x4_f32(
                    false, a[kk], false, b1, (short)0, acc1, false, false);
            }
        } else {
            // Edge tiles: unconditional loads from clamped addresses, then
            // branch-free zero-select (v_cndmask) — no EXEC manipulation.
#pragma unroll
            for (int kk = 0; kk < 32; ++kk) {
                const size_t co = (size_t)(4 * kk + 2 * hi) * planeHW;
                const float tA0 = pA[co], tA1 = pA[co + planeHW];
                const float tB0 = pB[co], tB1 = pB[co + planeHW];
                v2f b0, b1;
                b0.x = okA ? tA0 : 0.0f; b0.y = okA ? tA1 : 0.0f;
                b1.x = okB ? tB0 : 0.0f; b1.y = okB ? tB1 : 0.0f;
                acc0 = __builtin_amdgcn_wmma_f32_16x16x4_f32(
                    false, a[kk], false, b0, (short)0, acc0, false, false);
                acc1 = __builtin_amdgcn_wmma_f32_16x16x4_f32(
                    false, a[kk], false, b1, (short)0, acc1, false, false);
            }
        }

        // Lane (hi, n) holds D[m = 8*hi + r][n + 16*t] in acc_t[r].
        // Keep the band ox = j - m in [0, 8]; stage into LDS for coalesced stores.
#pragma unroll
        for (int r = 0; r < 8; ++r) {
            const int m   = 8 * hi + r;
            const int ox0 = ln - m;          // j = ln        (tile 0)
            const int ox1 = ln + 16 - m;     // j = ln + 16   (tile 1)
            if (ox0 >= 0 && ox0 <= 8) myLds[ox0 * 17 + m] = acc0[r];
            if (ox1 >= 0 && ox1 <= 8) myLds[ox1 * 17 + m] = acc1[r];
        }

        // Wave-private LDS patch: DS ops are in-order within a wave; wait for the
        // writes' data before cross-lane readback (no block barrier needed).
        asm volatile("s_wait_dscnt 0" ::: "memory");

        for (int e = lane; e < PS * 16; e += 32) {
            const int ox = e >> 4, m = e & 15;
            outOy[(size_t)ox * planeHW + m] = myLds[ox * 17 + m];
        }
    }
}

extern "C" void kernel_launch(void* const* d_in, const int* in_sizes, int n_in,
                              void* d_out, int out_size, void* d_ws, size_t ws_size,
                              hipStream_t stream) {
    const float* f0 = (const float*)d_in[0];
    const float* f1 = (const float*)d_in[1];
    float* out = (float*)d_out;
    // 8192 tiles (4 b x 128 h x 16 w-tiles), 8 waves (256 threads) per block.
    corr_wmma_f32_kernel<<<1024, 256, 0, stream>>>(f0, f1, out);
}